// RasterPoints_8091718386495
// MI455X (gfx1250) — compile-verified
//
#include <hip/hip_runtime.h>

// CDNA5 / gfx1250. wave32. This op is a pure 320MB streaming zero-fill plus a
// 2000-element scatter: memory-roofline bound (~13.7us @ 23.3 TB/s). No matrix
// math exists in the reference, so no WMMA is used; the gfx1250-specific path
// exercised here is the non-temporal (th:TH_STORE_NT) b128 global store, which
// avoids thrashing the 192MB L2 with a 320MB write-once stream.

typedef __attribute__((ext_vector_type(4))) float v4f;

// ---------------------------------------------------------------------------
// Kernel 1: zero-fill d_out with 16B non-temporal stores, grid-stride loop.
// Each wave32 store instruction streams 512 contiguous bytes.
// ---------------------------------------------------------------------------
__global__ void raster_zero_fill_nt(v4f* __restrict__ out16, size_t n16,
                                    float* __restrict__ out, size_t out_n) {
    const size_t stride = (size_t)gridDim.x * blockDim.x;
    size_t i = (size_t)blockIdx.x * blockDim.x + threadIdx.x;
    const v4f z = (v4f)(0.0f);
    for (; i < n16; i += stride) {
        __builtin_nontemporal_store(z, &out16[i]);
    }
    // Tail (out_n % 4 elements) — 80M is divisible by 4, but stay general.
    const size_t tail = out_n & (size_t)3;
    if (blockIdx.x == 0 && threadIdx.x < tail) {
        out[(n16 << 2) + threadIdx.x] = 0.0f;
    }
}

// ---------------------------------------------------------------------------
// Kernel 2: scatter 1.0f at the 2000 computed (bt,row,col,p) positions.
//   row = trunc(y / res[bt][0] + org[bt][0])
//   col = trunc(x / res[bt][1] + org[bt][1])
// Bounds-guarded to match JAX's drop-out-of-bounds scatter semantics.
// ---------------------------------------------------------------------------
__global__ void raster_scatter_ones(const float* __restrict__ x,
                                    const float* __restrict__ res,
                                    const float* __restrict__ org,
                                    float* __restrict__ out,
                                    int BT, int NP, int H, int W) {
    const int tid = blockIdx.x * blockDim.x + threadIdx.x;
    const int total = BT * NP;
    if (tid >= total) return;

    const int bt = tid / NP;
    const int p  = tid - bt * NP;

    const size_t xb = (size_t)bt * (size_t)(2 * NP);
    const float px = x[xb + 2 * p + 0];   // x-coordinate
    const float py = x[xb + 2 * p + 1];   // y-coordinate

    const float r0 = res[2 * bt + 0];
    const float r1 = res[2 * bt + 1];
    const float o0 = org[2 * bt + 0];
    const float o1 = org[2 * bt + 1];

    const int row = (int)(py / r0 + o0);  // C truncation == jnp astype(int32)
    const int col = (int)(px / r1 + o1);

    if (row >= 0 && row < H && col >= 0 && col < W) {
        const size_t idx =
            (((size_t)bt * (size_t)H + (size_t)row) * (size_t)W + (size_t)col)
                * (size_t)NP + (size_t)p;
        out[idx] = 1.0f;
    }
}

// ---------------------------------------------------------------------------
// Inputs (setup_inputs order):
//   d_in[0]: x          float32, B*T*NP*2   (=4000)
//   d_in[1]: resolution float32, B*T*2      (=400)
//   d_in[2]: origin     float32, B*T*2      (=400)
// Output: float32, B*T*H*W*NP (=80,000,000)
// ---------------------------------------------------------------------------
extern "C" void kernel_launch(void* const* d_in, const int* in_sizes, int n_in,
                              void* d_out, int out_size, void* d_ws, size_t ws_size,
                              hipStream_t stream) {
    const float* x   = (const float*)d_in[0];
    const float* res = (const float*)d_in[1];
    const float* org = (const float*)d_in[2];
    float* out = (float*)d_out;

    const int H = 200, W = 200;               // SDF_SHAPE
    const int BT = in_sizes[1] / 2;           // B*T = 200
    const int NP = in_sizes[0] / (BT * 2);    // 10

    // --- zero fill: 20M float4 stores, ~8 iterations per thread ---
    const size_t out_n = (size_t)out_size;
    const size_t n16   = out_n >> 2;
    const int threads  = 256;
    const size_t per_block = (size_t)threads * 8;
    int blocks = (int)((n16 + per_block - 1) / per_block);
    if (blocks < 1) blocks = 1;
    raster_zero_fill_nt<<<blocks, threads, 0, stream>>>((v4f*)out, n16, out, out_n);

    // --- scatter the 2000 ones ---
    const int total = BT * NP;
    const int sblocks = (total + 255) / 256;
    raster_scatter_ones<<<sblocks, 256, 0, stream>>>(x, res, org, out, BT, NP, H, W);
}